// HetGATLayer_3229815406929
// MI455X (gfx1250) — compile-verified
//
#include <hip/hip_runtime.h>
#include <math.h>

// ---------------------------------------------------------------------------
// HetGAT layer for MI455X (gfx1250, wave32). fp32 end-to-end using
// V_WMMA_F32_16X16X4_F32 for every GEMM-like stage.
// ---------------------------------------------------------------------------

typedef __attribute__((ext_vector_type(2))) float v2f;
typedef __attribute__((ext_vector_type(8))) float v8f;

#define BB 8
#define NN 1024
#define MM 128
#define FF 64     // F_OP == F_MA == OUT
#define FE 16
#define HID 128

__device__ __forceinline__ v8f wmma4(v2f a, v2f b, v8f c) {
  // D(16x16,f32) += A(16x4,f32) * B(4x16,f32)
  return __builtin_amdgcn_wmma_f32_16x16x4_f32(
      /*neg_a=*/false, a, /*neg_b=*/false, b,
      /*c_mod=*/(short)0, c, /*reuse_a=*/false, /*reuse_b=*/false);
}

// ---------------------------------------------------------------------------
// K1: C[rows x 64] = A[rows x 64] @ W[64 x 64]
// block = 128 threads (4 waves); wave w owns column tile w. grid.x = rows/16.
// ---------------------------------------------------------------------------
__global__ __launch_bounds__(128) void gemm_rows_64(
    const float* __restrict__ A, const float* __restrict__ W,
    float* __restrict__ C) {
  const int wv   = threadIdx.x >> 5;
  const int lane = threadIdx.x & 31;
  const int lm   = lane & 15;
  const int kh   = (lane >> 4) * 2;
  const int lh   = lane >> 4;
  const int row0 = blockIdx.x * 16;
  const int o    = wv * 16 + lm;
  const float* Arow = A + (size_t)(row0 + lm) * FF;

  v8f c = {};
#pragma unroll
  for (int k0 = 0; k0 < FF; k0 += 4) {
    v2f a, b;
    a.x = Arow[k0 + kh];
    a.y = Arow[k0 + kh + 1];
    b.x = W[(size_t)(k0 + kh) * FF + o];
    b.y = W[(size_t)(k0 + kh + 1) * FF + o];
    c = wmma4(a, b, c);
  }
#pragma unroll
  for (int r = 0; r < 8; ++r) {
    int m = r + 8 * lh;
    C[(size_t)(row0 + m) * FF + o] = c[r];
  }
}

// ---------------------------------------------------------------------------
// K2: per-node projections ha1/ha2/ham1 and row_mask (all(op_ma_adj==0))
// ---------------------------------------------------------------------------
__global__ void proj_node(const float* __restrict__ h,
                          const float* __restrict__ att_op,
                          const float* __restrict__ att_ma,
                          const int* __restrict__ opma,
                          float* __restrict__ ha1, float* __restrict__ ha2,
                          float* __restrict__ ham1, int* __restrict__ rowmask) {
  int idx = blockIdx.x * blockDim.x + threadIdx.x;
  if (idx >= BB * NN) return;
  const float* hr = h + (size_t)idx * FF;
  float s1 = 0.f, s2 = 0.f, s3 = 0.f;
#pragma unroll 8
  for (int o = 0; o < FF; ++o) {
    float v = hr[o];
    s1 += v * att_op[o];
    s2 += v * att_op[FF + o];
    s3 += v * att_ma[o];
  }
  ha1[idx] = s1; ha2[idx] = s2; ham1[idx] = s3;
  const int* ad = opma + (size_t)idx * MM;
  int any = 0;
#pragma unroll 8
  for (int m = 0; m < MM; ++m) any |= ad[m];
  rowmask[idx] = (any == 0) ? 1 : 0;
}

__global__ void proj_ma(const float* __restrict__ hma,
                        const float* __restrict__ att_ma,
                        float* __restrict__ hma2) {
  int idx = blockIdx.x * blockDim.x + threadIdx.x;
  if (idx >= BB * MM) return;
  const float* r = hma + (size_t)idx * FF;
  float s = 0.f;
#pragma unroll 8
  for (int o = 0; o < FF; ++o) s += r[o] * att_ma[FF + o];
  hma2[idx] = s;
}

// ---------------------------------------------------------------------------
// K3: op-attention.  One block = 16 query rows of one batch.
// Scores + masked softmax in LDS, then WMMA: att(16x1024) @ h_b(1024x64).
// Dynamic LDS: att[16][1028] + inv_sum[16]  (~66 KB)
// ---------------------------------------------------------------------------
__global__ __launch_bounds__(128) void op_attn(
    const float* __restrict__ h, const float* __restrict__ ha1,
    const float* __restrict__ ha2, const int* __restrict__ opadj,
    const int* __restrict__ maadj, const int* __restrict__ rowmask,
    float* __restrict__ hprime) {
  extern __shared__ float smem[];
  typedef float att_row_t[NN + 4];           // pad 4 dwords: stride%64 = 4
  att_row_t* att = (att_row_t*)smem;
  float* inv_sum = smem + 16 * (NN + 4);

  const int b  = blockIdx.y;
  const int i0 = blockIdx.x * 16;
  const int* opb = opadj + (size_t)b * NN * NN;
  const int* mab = maadj + (size_t)b * NN * NN;
  const float NEGINF = -__builtin_inff();

  // ---- phase 1: masked leaky-relu scores -------------------------------
  for (int t = threadIdx.x; t < 16 * NN; t += 128) {
    int r = t >> 10, j = t & (NN - 1);
    int i = i0 + r;
    int conn;
    if (rowmask[b * NN + i]) {
      conn = 1;
    } else {
      int adj = opb[(size_t)i * NN + j] + opb[(size_t)j * NN + i] +
                mab[(size_t)i * NN + j] + mab[(size_t)j * NN + i];
      conn = adj > 0;
    }
    float e = ha2[b * NN + i] + ha1[b * NN + j];
    e = e > 0.f ? e : 0.01f * e;
    att[r][j] = conn ? e : NEGINF;
  }
  __syncthreads();

  // ---- phase 2: per-row softmax (leave unnormalized exp; keep 1/sum) ---
  const int wv = threadIdx.x >> 5;
  const int lane = threadIdx.x & 31;
  for (int r = wv; r < 16; r += 4) {
    float mx = NEGINF;
    for (int j = lane; j < NN; j += 32) mx = fmaxf(mx, att[r][j]);
#pragma unroll
    for (int off = 16; off; off >>= 1) mx = fmaxf(mx, __shfl_xor(mx, off, 32));
    if (mx == NEGINF) mx = 0.f;
    float sum = 0.f;
    for (int j = lane; j < NN; j += 32) {
      float p = __expf(att[r][j] - mx);
      att[r][j] = p;
      sum += p;
    }
#pragma unroll
    for (int off = 16; off; off >>= 1) sum += __shfl_xor(sum, off, 32);
    if (lane == 0) inv_sum[r] = 1.f / fmaxf(sum, 1e-30f);
  }
  __syncthreads();

  // ---- phase 3: WMMA  hprime_op = softmax(att) @ h_b -------------------
  const int lm = lane & 15;
  const int kh = (lane >> 4) * 2;
  const int lh = lane >> 4;
  const int o  = wv * 16 + lm;
  const float* hb = h + (size_t)b * NN * FF;

  v8f c = {};
  for (int k0 = 0; k0 < NN; k0 += 4) {
    v2f a, bf;
    a.x  = att[lm][k0 + kh];
    a.y  = att[lm][k0 + kh + 1];
    bf.x = hb[(size_t)(k0 + kh) * FF + o];
    bf.y = hb[(size_t)(k0 + kh + 1) * FF + o];
    c = wmma4(a, bf, c);
  }
  float* hp = hprime + ((size_t)b * NN + i0) * (2 * FF);
#pragma unroll
  for (int r = 0; r < 8; ++r) {
    int m = r + 8 * lh;
    hp[(size_t)m * (2 * FF) + o] = c[r] * inv_sum[m];
  }
}

// ---------------------------------------------------------------------------
// K4: ma-attention, fully fused per (b,n): h_edge tile GEMM (WMMA) -> scores
// -> masked softmax over M=128 -> weighted reduction (+h residual).
// Static LDS ~48 KB.
// ---------------------------------------------------------------------------
__global__ __launch_bounds__(256) void ma_attn(
    const float* __restrict__ z, const float* __restrict__ wedge,
    const float* __restrict__ hma, const float* __restrict__ ham1,
    const float* __restrict__ hma2, const int* __restrict__ opma,
    const int* __restrict__ rowmask, const float* __restrict__ h,
    const float* __restrict__ attma, float* __restrict__ hprime) {
  __shared__ float zb[MM][FE + 1];    // pad: stride 17
  __shared__ float web[FE][FF];
  __shared__ float heb[MM][FF + 1];   // pad: stride 65
  __shared__ float sc[MM];
  __shared__ float red[2];
  __shared__ float psum[4][FF];

  const int b = blockIdx.y, n = blockIdx.x;
  const size_t bn = (size_t)b * NN + n;
  const float NEGINF = -__builtin_inff();

  const float* zrow = z + bn * MM * FE;
  for (int t = threadIdx.x; t < MM * FE; t += 256) zb[t >> 4][t & 15] = zrow[t];
  for (int t = threadIdx.x; t < FE * FF; t += 256) web[t >> 6][t & 63] = wedge[t];
  __syncthreads();

  // ---- WMMA: he(128x64) = z(128x16) @ we(16x64); wave w -> row tile w --
  const int wv = threadIdx.x >> 5;
  const int lane = threadIdx.x & 31;
  const int lm = lane & 15;
  const int kh = (lane >> 4) * 2;
  const int lh = lane >> 4;
  const int m0 = wv * 16;
  for (int ct = 0; ct < 4; ++ct) {
    v8f c = {};
#pragma unroll
    for (int k0 = 0; k0 < FE; k0 += 4) {
      v2f a, bf;
      a.x  = zb[m0 + lm][k0 + kh];
      a.y  = zb[m0 + lm][k0 + kh + 1];
      bf.x = web[k0 + kh][ct * 16 + lm];
      bf.y = web[k0 + kh + 1][ct * 16 + lm];
      c = wmma4(a, bf, c);
    }
#pragma unroll
    for (int r = 0; r < 8; ++r)
      heb[m0 + r + 8 * lh][ct * 16 + lm] = c[r];
  }
  __syncthreads();

  // ---- scores: e_ma[m] = leaky(ham1[n] + hma2[m] + he[m].am3) ----------
  const int rm = rowmask[bn];
  const int* ad = opma + bn * MM;
  const float* am3 = attma + 2 * FF;
  if (threadIdx.x < MM) {
    int m = threadIdx.x;
    float dot = 0.f;
#pragma unroll 8
    for (int o = 0; o < FF; ++o) dot += heb[m][o] * am3[o];
    float e = ham1[bn] + hma2[(size_t)b * MM + m] + dot;
    e = e > 0.f ? e : 0.01f * e;
    int conn = rm ? 1 : (ad[m] > 0);
    sc[m] = conn ? e : NEGINF;
  }
  __syncthreads();

  if (wv == 0) {
    float mx = fmaxf(fmaxf(sc[lane], sc[lane + 32]),
                     fmaxf(sc[lane + 64], sc[lane + 96]));
#pragma unroll
    for (int off = 16; off; off >>= 1) mx = fmaxf(mx, __shfl_xor(mx, off, 32));
    if (mx == NEGINF) mx = 0.f;
    if (lane == 0) red[0] = mx;
  }
  __syncthreads();
  const float mx = red[0];
  if (threadIdx.x < MM) sc[threadIdx.x] = __expf(sc[threadIdx.x] - mx);
  __syncthreads();
  if (wv == 0) {
    float s = sc[lane] + sc[lane + 32] + sc[lane + 64] + sc[lane + 96];
#pragma unroll
    for (int off = 16; off; off >>= 1) s += __shfl_xor(s, off, 32);
    if (lane == 0) red[1] = 1.f / fmaxf(s, 1e-30f);
  }
  __syncthreads();
  const float inv = red[1];

  // ---- weighted sum over m, split across 4 chunks ----------------------
  {
    int o = threadIdx.x & 63, ch = threadIdx.x >> 6;
    const float* hmb = hma + (size_t)b * MM * FF;
    float acc = 0.f;
    for (int m = ch * 32; m < ch * 32 + 32; ++m)
      acc += sc[m] * (hmb[(size_t)m * FF + o] + heb[m][o]);
    psum[ch][o] = acc;
  }
  __syncthreads();
  if (threadIdx.x < FF) {
    int o = threadIdx.x;
    float v = (psum[0][o] + psum[1][o] + psum[2][o] + psum[3][o]) * inv +
              h[bn * FF + o];
    hprime[bn * (2 * FF) + FF + o] = v;
  }
}

// ---------------------------------------------------------------------------
// K5: MLP  elu(hp@w1+b1) -> elu(@w2+b2) -> @wo+bo, row-mask zeroing.
// 16 rows / block, 8 waves, weights staged in dynamic LDS (~82 KB).
// ---------------------------------------------------------------------------
__global__ __launch_bounds__(256) void mlp(
    const float* __restrict__ hprime, const float* __restrict__ w1,
    const float* __restrict__ b1, const float* __restrict__ w2,
    const float* __restrict__ b2, const float* __restrict__ wo,
    const float* __restrict__ bo, const int* __restrict__ rowmask,
    float* __restrict__ out) {
  extern __shared__ float smem[];
  float* wb = smem;                               // [128][128] flat
  typedef float act_row_t[HID + 4];               // stride 132: stride%64 = 4
  act_row_t* a0 = (act_row_t*)(smem + HID * HID);
  act_row_t* a1 = (act_row_t*)(smem + HID * HID + 16 * (HID + 4));

  const int row0 = blockIdx.x * 16;
  const int wv = threadIdx.x >> 5;
  const int lane = threadIdx.x & 31;
  const int lm = lane & 15;
  const int kh = (lane >> 4) * 2;
  const int lh = lane >> 4;

  for (int t = threadIdx.x; t < 16 * HID; t += 256)
    a0[t >> 7][t & 127] = hprime[(size_t)row0 * HID + t];
  for (int t = threadIdx.x; t < HID * HID; t += 256) wb[t] = w1[t];
  __syncthreads();

  // ---- stage 1: a1 = elu(a0 @ w1 + b1), wave w -> col tile w -----------
  {
    int o = wv * 16 + lm;
    v8f c = {};
    for (int k0 = 0; k0 < HID; k0 += 4) {
      v2f a, bf;
      a.x  = a0[lm][k0 + kh];
      a.y  = a0[lm][k0 + kh + 1];
      bf.x = wb[(k0 + kh) * HID + o];
      bf.y = wb[(k0 + kh + 1) * HID + o];
      c = wmma4(a, bf, c);
    }
    float bias = b1[o];
#pragma unroll
    for (int r = 0; r < 8; ++r) {
      float v = c[r] + bias;
      a1[r + 8 * lh][o] = v > 0.f ? v : expm1f(v);
    }
  }
  __syncthreads();
  for (int t = threadIdx.x; t < HID * HID; t += 256) wb[t] = w2[t];
  __syncthreads();

  // ---- stage 2: a0 = elu(a1 @ w2 + b2) ---------------------------------
  {
    int o = wv * 16 + lm;
    v8f c = {};
    for (int k0 = 0; k0 < HID; k0 += 4) {
      v2f a, bf;
      a.x  = a1[lm][k0 + kh];
      a.y  = a1[lm][k0 + kh + 1];
      bf.x = wb[(k0 + kh) * HID + o];
      bf.y = wb[(k0 + kh + 1) * HID + o];
      c = wmma4(a, bf, c);
    }
    float bias = b2[o];
#pragma unroll
    for (int r = 0; r < 8; ++r) {
      float v = c[r] + bias;
      a0[r + 8 * lh][o] = v > 0.f ? v : expm1f(v);
    }
  }
  __syncthreads();
  for (int t = threadIdx.x; t < HID * FF; t += 256) wb[t] = wo[t];
  __syncthreads();

  // ---- stage 3: out = a0 @ wo + bo, mask rows; waves 0..3 --------------
  if (wv < 4) {
    int o = wv * 16 + lm;
    v8f c = {};
    for (int k0 = 0; k0 < HID; k0 += 4) {
      v2f a, bf;
      a.x  = a0[lm][k0 + kh];
      a.y  = a0[lm][k0 + kh + 1];
      bf.x = wb[(k0 + kh) * FF + o];
      bf.y = wb[(k0 + kh + 1) * FF + o];
      c = wmma4(a, bf, c);
    }
    float bias = bo[o];
#pragma unroll
    for (int r = 0; r < 8; ++r) {
      int m = r + 8 * lh;
      float v = c[r] + bias;
      if (rowmask[row0 + m]) v = 0.f;
      out[(size_t)(row0 + m) * FF + o] = v;
    }
  }
}

// ---------------------------------------------------------------------------
extern "C" void kernel_launch(void* const* d_in, const int* in_sizes, int n_in,
                              void* d_out, int out_size, void* d_ws,
                              size_t ws_size, hipStream_t stream) {
  (void)in_sizes; (void)n_in; (void)out_size; (void)ws_size;
  const float* x      = (const float*)d_in[0];
  const float* y      = (const float*)d_in[1];
  const float* z      = (const float*)d_in[2];
  const int*   op_adj = (const int*)d_in[3];
  const int*   ma_adj = (const int*)d_in[4];
  const int*   op_ma  = (const int*)d_in[5];
  const float* Wop    = (const float*)d_in[6];
  const float* Wma    = (const float*)d_in[7];
  const float* Wedge  = (const float*)d_in[8];
  const float* att_op = (const float*)d_in[9];
  const float* att_ma = (const float*)d_in[10];
  const float* w1     = (const float*)d_in[11];
  const float* b1     = (const float*)d_in[12];
  const float* w2     = (const float*)d_in[13];
  const float* b2     = (const float*)d_in[14];
  const float* wo     = (const float*)d_in[15];
  const float* bo     = (const float*)d_in[16];
  float* out = (float*)d_out;

  float* ws     = (float*)d_ws;
  float* h      = ws;                    // 8*1024*64   = 524288
  float* hma    = h + BB * NN * FF;      // 8*128*64    =  65536
  float* ha1    = hma + BB * MM * FF;    // 8192
  float* ha2    = ha1 + BB * NN;         // 8192
  float* ham1   = ha2 + BB * NN;         // 8192
  float* hma2   = ham1 + BB * NN;        // 1024
  float* hprime = hma2 + BB * MM;        // 8*1024*128  = 1048576
  int*   rmask  = (int*)(hprime + (size_t)BB * NN * 2 * FF);  // 8192 ints

  // h = x @ Wop ; hma = y @ Wma
  gemm_rows_64<<<dim3((BB * NN) / 16), dim3(128), 0, stream>>>(x, Wop, h);
  gemm_rows_64<<<dim3((BB * MM) / 16), dim3(128), 0, stream>>>(y, Wma, hma);
  // projections + row mask
  proj_node<<<dim3((BB * NN) / 256), dim3(256), 0, stream>>>(
      h, att_op, att_ma, op_ma, ha1, ha2, ham1, rmask);
  proj_ma<<<dim3((BB * MM) / 256), dim3(256), 0, stream>>>(hma, att_ma, hma2);
  // op attention (writes hprime[..., :64])
  size_t sm2 = (16 * (NN + 4) + 16) * sizeof(float);
  op_attn<<<dim3(NN / 16, BB), dim3(128), sm2, stream>>>(
      h, ha1, ha2, op_adj, ma_adj, rmask, hprime);
  // ma attention fused with h_edge GEMM (writes hprime[..., 64:])
  ma_attn<<<dim3(NN, BB), dim3(256), 0, stream>>>(
      z, Wedge, hma, ham1, hma2, op_ma, rmask, h, att_ma, hprime);
  // MLP head
  size_t sm4 = (HID * HID + 2 * 16 * (HID + 4)) * sizeof(float);
  mlp<<<dim3((BB * NN) / 16), dim3(256), sm4, stream>>>(
      hprime, w1, b1, w2, b2, wo, bo, rmask, out);
}